// GNNEncoder_61899068670163
// MI455X (gfx1250) — compile-verified
//
#include <hip/hip_runtime.h>
#include <hip/hip_bf16.h>

typedef float v2f __attribute__((ext_vector_type(2)));
typedef float v8f __attribute__((ext_vector_type(8)));

#define NN 100000      // nodes
#define NE 1600000     // edges

// ---------------------------------------------------------------------------
// Degree: f32 atomic histogram over dst, then in-place 1/max(deg,1)
// ---------------------------------------------------------------------------
__global__ __launch_bounds__(256) void degree_kernel(const long long* __restrict__ dst,
                                                     float* __restrict__ deg, int nEdges) {
    int t = blockIdx.x * 256 + threadIdx.x;
    if (t < nEdges) atomicAdd(&deg[(int)dst[t]], 1.0f);
}

__global__ __launch_bounds__(256) void invdeg_kernel(float* __restrict__ deg, int n) {
    int t = blockIdx.x * 256 + threadIdx.x;
    if (t < n) deg[t] = 1.0f / fmaxf(deg[t], 1.0f);
}

// ---------------------------------------------------------------------------
// Zero-pad a matrix [rIn x cIn] -> [rOut x cOut]
// ---------------------------------------------------------------------------
__global__ __launch_bounds__(256) void pad_mat(const float* __restrict__ in, float* __restrict__ out,
                                               int rIn, int cIn, int rOut, int cOut) {
    int t = blockIdx.x * 256 + threadIdx.x;
    if (t >= rOut * cOut) return;
    int r = t / cOut, c = t % cOut;
    out[t] = (r < rIn && c < cIn) ? in[r * cIn + c] : 0.0f;
}

// ---------------------------------------------------------------------------
// Edge scatter-add: one wave (32 lanes) per edge; lane c moves features
// [4c, 4c+4) as a float4 (GLOBAL_LOAD_B128) + 4 f32 atomics into agg[dst].
// Feature matrix (<=51MB) is L2-resident; atomics resolve in L2.
// blockDim=256 -> 8 edges/block.
// ---------------------------------------------------------------------------
__global__ __launch_bounds__(256) void scatter_add(const long long* __restrict__ src,
                                                   const long long* __restrict__ dst,
                                                   const float* __restrict__ h,
                                                   float* __restrict__ agg,
                                                   int nEdges, int D) {
    int lane = threadIdx.x & 31;
    int e = blockIdx.x * 8 + (threadIdx.x >> 5);
    int chunks = D >> 2;
    if (e >= nEdges || lane >= chunks) return;
    long long s = src[e], d = dst[e];
    const float4 v = *(const float4*)(h + s * (long long)D + lane * 4);
    float* p = agg + d * (long long)D + lane * 4;
    atomicAdd(p + 0, v.x);
    atomicAdd(p + 1, v.y);
    atomicAdd(p + 2, v.z);
    atomicAdd(p + 3, v.w);
}

// ---------------------------------------------------------------------------
// Fused SAGE layer GEMM on the matrix pipe:
//   out = relu?( (agg * inv_deg[row]) @ Wl + h @ Wr + bias )
// One wave per 16x16 C tile; blockDim = 32 * (N/16); gridDim.x = M/16.
// K-loop accumulates two V_WMMA_F32_16X16X4_F32 per step into one v8f acc.
// A layout (ISA 7.12.2, 32-bit A 16x4): lane<16 -> K={0,1}, lane>=16 -> K={2,3},
// row M = lane&15. B (4x16): row-striped across lanes, mirrored halves.
// C/D: VGPR r -> M = r + 8*(lane>>4), N = lane&15.
// ---------------------------------------------------------------------------
__global__ __launch_bounds__(256) void sage_gemm(const float* __restrict__ agg,
                                                 const float* __restrict__ h,
                                                 const float* __restrict__ inv_deg,
                                                 const float* __restrict__ Wl,
                                                 const float* __restrict__ Wr,
                                                 const float* __restrict__ bias,
                                                 float* __restrict__ out,
                                                 int N, int K, int relu) {
    const int wave = threadIdx.x >> 5;
    const int lane = threadIdx.x & 31;
    const int half = lane >> 4;
    const int l16  = lane & 15;

    const int mBase = blockIdx.x * 16;
    const int m = mBase + l16;            // A row handled by this lane
    const int n = wave * 16 + l16;        // B/C column handled by this lane

    const float scale = inv_deg[m];
    const float* aggRow = agg + (size_t)m * K;
    const float* hRow   = h   + (size_t)m * K;

    v8f c = {};
    for (int k0 = 0; k0 < K; k0 += 4) {
        const int ka = k0 + 2 * half;     // this lane's two K slots
        v2f aA, aH, bL, bR;
        aA.x = aggRow[ka] * scale;  aA.y = aggRow[ka + 1] * scale;
        aH.x = hRow[ka];            aH.y = hRow[ka + 1];
        bL.x = Wl[(size_t)ka * N + n];  bL.y = Wl[(size_t)(ka + 1) * N + n];
        bR.x = Wr[(size_t)ka * N + n];  bR.y = Wr[(size_t)(ka + 1) * N + n];
        c = __builtin_amdgcn_wmma_f32_16x16x4_f32(false, aA, false, bL, (short)0, c, false, false);
        c = __builtin_amdgcn_wmma_f32_16x16x4_f32(false, aH, false, bR, (short)0, c, false, false);
    }

    const float bv = bias[n];
    float* outCol = out + (size_t)mBase * N + n;
#pragma unroll
    for (int r = 0; r < 8; ++r) {
        int row = r + 8 * half;
        float v = c[r] + bv;
        if (relu) v = fmaxf(v, 0.0f);
        outCol[(size_t)row * N] = v;
    }
}

// ---------------------------------------------------------------------------
// Launch. Inputs: x, edge_index(int64, [2,NE]), Wl1,Wr1,b1, Wl2,Wr2,b2, Wl3,Wr3,b3
// Output: [NN x 64] f32.
// ---------------------------------------------------------------------------
extern "C" void kernel_launch(void* const* d_in, const int* in_sizes, int n_in,
                              void* d_out, int out_size, void* d_ws, size_t ws_size,
                              hipStream_t stream) {
    const float*      x   = (const float*)d_in[0];
    const long long*  ei  = (const long long*)d_in[1];
    const float* Wl1 = (const float*)d_in[2];
    const float* Wr1 = (const float*)d_in[3];
    const float* b1  = (const float*)d_in[4];
    const float* Wl2 = (const float*)d_in[5];
    const float* Wr2 = (const float*)d_in[6];
    const float* b2  = (const float*)d_in[7];
    const float* Wl3 = (const float*)d_in[8];
    const float* Wr3 = (const float*)d_in[9];
    const float* b3  = (const float*)d_in[10];

    const int nNodes = NN;
    const int nEdges = in_sizes[1] / 2;
    const long long* src = ei;
    const long long* dst = ei + nEdges;

    const int H1 = 128, H2P = 80, DOUT = 64;   // H2 padded 76 -> 80

    // ---- workspace carve-up (floats) ----
    float* ws   = (float*)d_ws;
    float* deg  = ws;                                 // NN            (becomes inv_deg)
    float* agg  = deg  + nNodes;                      // NN*128 (reused every layer)
    float* h1   = agg  + (size_t)nNodes * 128;        // NN*128
    float* h2   = h1   + (size_t)nNodes * 128;        // NN*80
    float* Wl2p = h2   + (size_t)nNodes * H2P;        // 128*80
    float* Wr2p = Wl2p + 128 * H2P;                   // 128*80
    float* b2p  = Wr2p + 128 * H2P;                   // 80
    float* Wl3p = b2p  + H2P;                         // 80*64
    float* Wr3p = Wl3p + H2P * DOUT;                  // 80*64

    // ---- degree / inv_deg ----
    hipMemsetAsync(deg, 0, (size_t)nNodes * sizeof(float), stream);
    degree_kernel<<<(nEdges + 255) / 256, 256, 0, stream>>>(dst, deg, nEdges);
    invdeg_kernel<<<(nNodes + 255) / 256, 256, 0, stream>>>(deg, nNodes);

    // ---- pad layer-2/3 params (pad lanes yield exact zeros through ReLU) ----
    pad_mat<<<(128 * H2P + 255) / 256, 256, 0, stream>>>(Wl2, Wl2p, 128, 76, 128, H2P);
    pad_mat<<<(128 * H2P + 255) / 256, 256, 0, stream>>>(Wr2, Wr2p, 128, 76, 128, H2P);
    pad_mat<<<(H2P + 255) / 256, 256, 0, stream>>>(b2, b2p, 1, 76, 1, H2P);
    pad_mat<<<(H2P * DOUT + 255) / 256, 256, 0, stream>>>(Wl3, Wl3p, 76, DOUT, H2P, DOUT);
    pad_mat<<<(H2P * DOUT + 255) / 256, 256, 0, stream>>>(Wr3, Wr3p, 76, DOUT, H2P, DOUT);

    const int edgeBlocks = (nEdges + 7) / 8;   // 8 edges per 256-thread block
    const int mTiles = nNodes / 16;            // 6250

    // ---- layer 1: D=128, N=128, K=128, relu ----
    hipMemsetAsync(agg, 0, (size_t)nNodes * 128 * sizeof(float), stream);
    scatter_add<<<edgeBlocks, 256, 0, stream>>>(src, dst, x, agg, nEdges, 128);
    sage_gemm<<<mTiles, 8 * 32, 0, stream>>>(agg, x, deg, Wl1, Wr1, b1, h1, 128, 128, 1);

    // ---- layer 2: D=128, N=80(padded 76), K=128, relu ----
    hipMemsetAsync(agg, 0, (size_t)nNodes * 128 * sizeof(float), stream);
    scatter_add<<<edgeBlocks, 256, 0, stream>>>(src, dst, h1, agg, nEdges, 128);
    sage_gemm<<<mTiles, 5 * 32, 0, stream>>>(agg, h1, deg, Wl2p, Wr2p, b2p, h2, H2P, 128, 1);

    // ---- layer 3: D=80, N=64, K=80, no relu ----
    hipMemsetAsync(agg, 0, (size_t)nNodes * H2P * sizeof(float), stream);
    scatter_add<<<edgeBlocks, 256, 0, stream>>>(src, dst, h2, agg, nEdges, H2P);
    sage_gemm<<<mTiles, 4 * 32, 0, stream>>>(agg, h2, deg, Wl3p, Wr3p, b3, (float*)d_out, DOUT, H2P, 0);
}